// hardTripletLoss_3908420239845
// MI455X (gfx1250) — compile-verified
//
#include <hip/hip_runtime.h>
#include <math.h>

typedef float v2f __attribute__((ext_vector_type(2)));
typedef float v8f __attribute__((ext_vector_type(8)));

#define MARGIN 0.3f
#define EPS2 1e-12f

// ---------------- Kernel 1: row squared-norms + init of ap/an buffers --------
__global__ __launch_bounds__(256)
void tri_prep_kernel(const float* __restrict__ X, float* __restrict__ sq,
                     unsigned* __restrict__ apBits, unsigned* __restrict__ anBits,
                     int n, int d) {
    const int wave = threadIdx.x >> 5;
    const int lane = threadIdx.x & 31;
    const int row  = blockIdx.x * 8 + wave;
    if (row >= n) return;
    const float* p = X + (size_t)row * d;
    float s = 0.0f;
    for (int k = lane * 4; k < d; k += 32 * 4) {
        float4 v = *(const float4*)(p + k);
        s += v.x * v.x + v.y * v.y + v.z * v.z + v.w * v.w;
    }
    #pragma unroll
    for (int m = 16; m >= 1; m >>= 1) s += __shfl_xor(s, m, 32);
    if (lane == 0) {
        sq[row]     = s;
        apBits[row] = 0u;           // dist >= 1e-6 > 0, uint-bit max == float max
        anBits[row] = 0x7F800000u;  // +inf
    }
}

// ---------------- Kernel 2: fused WMMA Gram tiles + hardest pos/neg ---------
__global__ __launch_bounds__(256)
void tri_gram_kernel(const float* __restrict__ X, const long long* __restrict__ tgt,
                     const float* __restrict__ sq,
                     unsigned* __restrict__ apBits, unsigned* __restrict__ anBits,
                     int d) {
    constexpr int BM = 64, BN = 128, KC = 64, LDA = KC + 4; // 68-float stride (16B aligned, bank-safe)
    __shared__ float As[BM * LDA];
    __shared__ float Bs[BN * LDA];

    const int tid  = threadIdx.x;
    const int wave = tid >> 5;     // 0..7
    const int lane = tid & 31;
    const int wr   = wave >> 1;    // row-tile 0..3 (16 rows each)
    const int wc   = wave & 1;     // col-half 0..1 (64 cols each)
    const int hi   = lane >> 4;    // half-wave id
    const int l15  = lane & 15;
    const int koff = hi * 2;       // K sub-offset per ISA A/B f32 layout

    const int rowBase = blockIdx.x * BM;
    const int colBase = blockIdx.y * BN;

    v8f acc[4];
    #pragma unroll
    for (int c = 0; c < 4; ++c) acc[c] = (v8f){0.f,0.f,0.f,0.f,0.f,0.f,0.f,0.f};

    for (int kb = 0; kb < d; kb += KC) {
        // Stage A tile: BM x KC  (16 float4 per row)
        for (int i = tid; i < BM * (KC / 4); i += 256) {
            const int r = i >> 4, c4 = i & 15;
            float4 v = *(const float4*)(X + (size_t)(rowBase + r) * d + kb + c4 * 4);
            *(float4*)(&As[r * LDA + c4 * 4]) = v;
        }
        // Stage B tile: BN x KC
        for (int i = tid; i < BN * (KC / 4); i += 256) {
            const int r = i >> 4, c4 = i & 15;
            float4 v = *(const float4*)(X + (size_t)(colBase + r) * d + kb + c4 * 4);
            *(float4*)(&Bs[r * LDA + c4 * 4]) = v;
        }
        __syncthreads();

        const float* aRow = &As[(wr * 16 + l15) * LDA];
        const float* bRow = &Bs[(wc * 64 + l15) * LDA];
        #pragma unroll
        for (int k = 0; k < KC; k += 4) {
            v2f a = *(const v2f*)(aRow + k + koff);
            #pragma unroll
            for (int c = 0; c < 4; ++c) {
                v2f b = *(const v2f*)(bRow + c * 16 * LDA + k + koff);
                acc[c] = __builtin_amdgcn_wmma_f32_16x16x4_f32(
                    false, a, false, b, (short)0, acc[c], false, false);
            }
        }
        __syncthreads();
    }

    // ---- Epilogue: distances + masked row max/min ----
    const int j0 = colBase + wc * 64;
    float sqj[4]; long long tj[4];
    #pragma unroll
    for (int c = 0; c < 4; ++c) {
        const int j = j0 + c * 16 + l15;
        sqj[c] = sq[j];
        tj[c]  = tgt[j];
    }
    const int iBase = rowBase + wr * 16 + hi * 8;
    #pragma unroll
    for (int r = 0; r < 8; ++r) {
        const int i = iBase + r;
        const float sqi   = sq[i];
        const long long ti = tgt[i];
        float mx = -1.0f;      // "no positive seen" sentinel (real dists > 0)
        float mn = INFINITY;   // "no negative seen" == init value, atomic is a no-op
        #pragma unroll
        for (int c = 0; c < 4; ++c) {
            const float g    = acc[c][r];
            const float d2   = sqi + sqj[c] - 2.0f * g;
            const float dist = sqrtf(fmaxf(d2, EPS2));
            if (ti == tj[c]) mx = fmaxf(mx, dist);
            else             mn = fminf(mn, dist);
        }
        // reduce across the 16 lanes (columns) of this half-wave
        #pragma unroll
        for (int m = 8; m >= 1; m >>= 1) {
            mx = fmaxf(mx, __shfl_xor(mx, m, 32));
            mn = fminf(mn, __shfl_xor(mn, m, 32));
        }
        if (l15 == 0) {
            if (mx >= 0.0f) atomicMax(&apBits[i], __float_as_uint(mx));
            atomicMin(&anBits[i], __float_as_uint(mn));
        }
    }
}

// ---------------- Kernel 3: hinge-mean reduction ----------------------------
__global__ __launch_bounds__(1024)
void tri_loss_kernel(const unsigned* __restrict__ apBits, const unsigned* __restrict__ anBits,
                     float* __restrict__ out, int n) {
    __shared__ float part[32];
    float s = 0.0f;
    for (int i = threadIdx.x; i < n; i += 1024) {
        const float ap = __uint_as_float(apBits[i]);
        const float an = __uint_as_float(anBits[i]);
        const float v  = ap - an + MARGIN;   // -inf handled: relu -> 0
        s += (v > 0.0f) ? v : 0.0f;
    }
    #pragma unroll
    for (int m = 16; m >= 1; m >>= 1) s += __shfl_xor(s, m, 32);
    const int wave = threadIdx.x >> 5, lane = threadIdx.x & 31;
    if (lane == 0) part[wave] = s;
    __syncthreads();
    if (wave == 0) {
        float t = part[lane];
        #pragma unroll
        for (int m = 16; m >= 1; m >>= 1) t += __shfl_xor(t, m, 32);
        if (lane == 0) out[0] = t / (float)n;
    }
}

extern "C" void kernel_launch(void* const* d_in, const int* in_sizes, int n_in,
                              void* d_out, int out_size, void* d_ws, size_t ws_size,
                              hipStream_t stream) {
    const float*     X   = (const float*)d_in[0];
    const long long* tgt = (const long long*)d_in[1];
    const int n = in_sizes[1];            // 8192
    const int d = in_sizes[0] / n;        // 256

    float*    sq     = (float*)d_ws;
    unsigned* apBits = (unsigned*)((char*)d_ws + (size_t)n * 4);
    unsigned* anBits = (unsigned*)((char*)d_ws + (size_t)n * 8);

    tri_prep_kernel<<<(n + 7) / 8, 256, 0, stream>>>(X, sq, apBits, anBits, n, d);

    dim3 grid(n / 64, n / 128);
    tri_gram_kernel<<<grid, 256, 0, stream>>>(X, tgt, sq, apBits, anBits, d);

    tri_loss_kernel<<<1, 1024, 0, stream>>>(apBits, anBits, (float*)d_out, n);
}